// MultiHeadAttention_16441134809117
// MI455X (gfx1250) — compile-verified
//
#include <hip/hip_runtime.h>
#include <math.h>

typedef float v2f __attribute__((ext_vector_type(2)));
typedef float v8f __attribute__((ext_vector_type(8)));
typedef unsigned int v4u __attribute__((ext_vector_type(4)));
typedef int v4i __attribute__((ext_vector_type(4)));
typedef int v8i __attribute__((ext_vector_type(8)));

#define HEADS  12
#define DMODEL 768
#define HDIM   64
#define BATCH  2
#define SEQ    2048

// ---- WMMA: D = A(16x4 f32) * B(4x16 f32) + C, full fp32 ----
__device__ __forceinline__ v8f wmma_f32(v2f a, v2f b, v8f c) {
  return __builtin_amdgcn_wmma_f32_16x16x4_f32(false, a, false, b, (short)0, c,
                                               false, false);
}

// ---- Tensor Data Mover (gfx1250 async tensor) with fallback ----
#if defined(__HIP_DEVICE_COMPILE__) &&                     \
    __has_builtin(__builtin_amdgcn_tensor_load_to_lds) &&  \
    __has_builtin(__builtin_amdgcn_s_wait_tensorcnt)
#define USE_TDM 1
#else
#define USE_TDM 0
#endif

#if USE_TDM
// 2D tile load, data_size = 4B.  Dims/strides in element (float) units.
// g1w0extra carries pad_enable/pad_interval/pad_amount bits.
__device__ __forceinline__ void tdm_load_2d(unsigned lds_addr, const void* gptr,
                                            unsigned tdim0, unsigned tdim1,
                                            unsigned tile0, unsigned tile1,
                                            unsigned long long stride0,
                                            unsigned g1w0extra) {
  unsigned long long ga = (unsigned long long)(uintptr_t)gptr;
  v4u g0;
  g0.x = 1u;                                        // count=1, user mode
  g0.y = lds_addr;                                  // LDS byte address
  g0.z = (unsigned)(ga & 0xFFFFFFFFu);              // global_addr[31:0]
  g0.w = (unsigned)((ga >> 32) & 0x1FFFFFFu) | (2u << 30);  // [56:32] | type=2
  v8i g1;
  g1[0] = (int)((2u << 16) | g1w0extra);            // data_size=4B code 2
  g1[1] = (int)((tdim0 & 0xFFFFu) << 16);           // tensor_dim0[15:0]
  g1[2] = (int)((tdim0 >> 16) | ((tdim1 & 0xFFFFu) << 16));
  g1[3] = (int)((tdim1 >> 16) | (tile0 << 16));     // tile_dim0
  g1[4] = (int)(tile1 & 0xFFFFu);                   // tile_dim1 (tile_dim2=0)
  g1[5] = (int)(stride0 & 0xFFFFFFFFu);             // tensor_dim0_stride
  g1[6] = (int)((stride0 >> 32) & 0xFFFFu);
  g1[7] = 0;
  v4i gz = {};
#if __clang_major__ >= 23
  v8i gz8 = {};
  __builtin_amdgcn_tensor_load_to_lds(g0, g1, gz, gz, gz8, 0);
#else
  __builtin_amdgcn_tensor_load_to_lds(g0, g1, gz, gz, 0);
#endif
}
#endif

// =====================================================================
// Projection: Y = X @ W^T + bias.  64x128 tile per WG, 256 thr (8 waves),
// each wave = one 16-col tile x 4 row-tiles sharing every W float2 load
// (4x less W L2 traffic).  headSplit: 0 -> [M,768]; 1 -> [B,H,S,64].
// =====================================================================
#define APSTR 772  // 768+4 pad: (772m+c)%64=(4m+c)%64 -> conflict-free

__global__ __launch_bounds__(256)
void mha_proj_kernel(const float* __restrict__ X, const float* __restrict__ W,
                     const float* __restrict__ bias, float* __restrict__ Y,
                     int headSplit) {
  extern __shared__ float lds_a[];  // 64 * APSTR floats

  const int tid   = threadIdx.x;
  const int mbase = blockIdx.x * 64;
  const int nbase = blockIdx.y * 128;

  for (int i = tid; i < 64 * (DMODEL / 4); i += 256) {
    int r = i / (DMODEL / 4);
    int c = (i % (DMODEL / 4)) * 4;
    float4 v = *(const float4*)(X + (size_t)(mbase + r) * DMODEL + c);
    float* dst = &lds_a[r * APSTR + c];
    dst[0] = v.x; dst[1] = v.y; dst[2] = v.z; dst[3] = v.w;
  }
  __syncthreads();

  const int wave  = tid >> 5, lane = tid & 31;
  const int m     = lane & 15;
  const int koff  = (lane >> 4) * 2;
  const int rbase = (lane >> 4) * 8;
  const int ncol  = nbase + wave * 16 + m;
  const float* wrow = W + (size_t)ncol * DMODEL;

  v8f acc[4] = {{}, {}, {}, {}};
  for (int kk = 0; kk < DMODEL; kk += 4) {
    v2f b = *(const v2f*)(wrow + kk + koff);  // shared by 4 row-tiles
#pragma unroll
    for (int t = 0; t < 4; ++t) {
      v2f a = *(const v2f*)(&lds_a[(t * 16 + m) * APSTR + kk + koff]);
      acc[t] = wmma_f32(a, b, acc[t]);
    }
  }

  const float bv = bias[ncol];
  if (headSplit) {
    const int h = ncol / HDIM, dh = ncol % HDIM;
#pragma unroll
    for (int t = 0; t < 4; ++t)
#pragma unroll
      for (int r = 0; r < 8; ++r) {
        int gm = mbase + t * 16 + rbase + r;
        int b_ = gm / SEQ, s_ = gm % SEQ;
        Y[(((size_t)(b_ * HEADS + h)) * SEQ + s_) * HDIM + dh] = acc[t][r] + bv;
      }
  } else {
#pragma unroll
    for (int t = 0; t < 4; ++t)
#pragma unroll
      for (int r = 0; r < 8; ++r) {
        int gm = mbase + t * 16 + rbase + r;
        Y[(size_t)gm * DMODEL + ncol] = acc[t][r] + bv;
      }
  }
}

// =====================================================================
// Fused attention: one WG = (b, h, 32 query rows), 256 thr (8 waves).
// 32x2048 fp32 prob panel in LDS (257KB of CDNA5's 320KB/WGP):
// weights written once, never re-read; K/V read once per 32 rows.
// =====================================================================
#define PSTR 2052
#define QSTR 68

__global__ __launch_bounds__(256)
void mha_attn_kernel(const float* __restrict__ Q, const float* __restrict__ K,
                     const float* __restrict__ V, const float* __restrict__ mask,
                     float* __restrict__ weights, float* __restrict__ ctx) {
  extern __shared__ float smem[];
  float* lds_p   = smem;                  // 32 * PSTR
  float* lds_q   = lds_p + 32 * PSTR;     // 32 * QSTR (reused as ctx partials)
  float* lds_m   = lds_q + 32 * QSTR;     // SEQ
  float* lds_red = lds_m + SEQ;           // 32 * 8

  const int tid  = threadIdx.x;
  const int wg   = blockIdx.x;
  const int qblk = wg & (SEQ / 32 - 1);
  const int h    = (wg >> 6) % HEADS;
  const int b    = wg / ((SEQ / 32) * HEADS);

  const size_t headoff = ((size_t)(b * HEADS + h)) * SEQ * HDIM;
  const float* Qh = Q + headoff;
  const float* Kh = K + headoff;
  const float* Vh = V + headoff;

  // ---- stage Q tile (32x64 -> stride-68 padded) + mask row ----
#if USE_TDM
  if ((tid >> 5) == 0) {
    // hardware-padded tile: 4 dwords of pad after every 64 dwords == QSTR
    const unsigned padbits = (1u << 20) | (5u << 22) | (3u << 25);
    tdm_load_2d((unsigned)(uintptr_t)lds_q, Qh + (size_t)qblk * 32 * HDIM,
                HDIM, SEQ, HDIM, 32, HDIM, padbits);
    tdm_load_2d((unsigned)(uintptr_t)lds_m, mask + (size_t)b * SEQ,
                SEQ, 1, SEQ, 1, SEQ, 0u);
    __builtin_amdgcn_s_wait_tensorcnt(0);
  }
#else
  for (int i = tid; i < 32 * HDIM; i += 256) {
    int r = i >> 6, c = i & 63;
    lds_q[r * QSTR + c] = Qh[(size_t)(qblk * 32 + r) * HDIM + c];
  }
  for (int i = tid; i < SEQ; i += 256) lds_m[i] = mask[(size_t)b * SEQ + i];
#endif
  __syncthreads();

  const int wave  = tid >> 5, lane = tid & 31;
  const int m     = lane & 15;
  const int koff  = (lane >> 4) * 2;
  const int rbase = (lane >> 4) * 8;

  // ---- scores: each K block feeds 2 row-tiles (shared B operand) ----
  for (int kb = wave; kb < SEQ / 16; kb += 8) {
    if (kb + 8 < SEQ / 16)
      __builtin_prefetch(Kh + (size_t)((kb + 8) * 16 + m) * HDIM, 0, 1);
    v8f acc0 = {}, acc1 = {};
    const float* krow = Kh + (size_t)(kb * 16 + m) * HDIM;  // lane = B col n
    for (int kk = 0; kk < HDIM; kk += 4) {
      v2f bb = *(const v2f*)(krow + kk + koff);
      v2f a0 = *(const v2f*)(&lds_q[m * QSTR + kk + koff]);
      v2f a1 = *(const v2f*)(&lds_q[(m + 16) * QSTR + kk + koff]);
      acc0 = wmma_f32(a0, bb, acc0);
      acc1 = wmma_f32(a1, bb, acc1);
    }
    const int col = kb * 16 + m;
    const bool msk = (lds_m[col] != 0.0f);
#pragma unroll
    for (int r = 0; r < 8; ++r) {
      float s0 = acc0[r] * 0.125f, s1 = acc1[r] * 0.125f;
      if (msk) { s0 = -INFINITY; s1 = -INFINITY; }
      lds_p[(rbase + r) * PSTR + col]      = s0;
      lds_p[(16 + rbase + r) * PSTR + col] = s1;
    }
  }
  __syncthreads();

  // ---- softmax: 8 threads per row over 2048 cols ----
  const int srow = tid >> 3, part = tid & 7;
  float lmax = -INFINITY;
  for (int j = 0; j < SEQ / 8; ++j)
    lmax = fmaxf(lmax, lds_p[srow * PSTR + part + 8 * j]);
  lds_red[srow * 8 + part] = lmax;
  __syncthreads();
  float rmax = lds_red[srow * 8];
#pragma unroll
  for (int t = 1; t < 8; ++t) rmax = fmaxf(rmax, lds_red[srow * 8 + t]);
  __syncthreads();

  float lsum = 0.0f;
  for (int j = 0; j < SEQ / 8; ++j) {
    int idx = srow * PSTR + part + 8 * j;
    float e = __expf(lds_p[idx] - rmax);
    lds_p[idx] = e;
    lsum += e;
  }
  lds_red[srow * 8 + part] = lsum;
  __syncthreads();
  float rsum = 0.0f;
#pragma unroll
  for (int t = 0; t < 8; ++t) rsum += lds_red[srow * 8 + t];
  const float rinv = 1.0f / rsum;

  float* wout =
      weights + (((size_t)(b * HEADS + h)) * SEQ + qblk * 32 + srow) * SEQ;
  for (int j = 0; j < SEQ / 8; ++j) {
    int c = part + 8 * j;
    float p = lds_p[srow * PSTR + c] * rinv;
    lds_p[srow * PSTR + c] = p;
    wout[c] = p;  // weights output: written exactly once
  }
  __syncthreads();

  // ---- ctx = P(32x2048) @ V(2048x64); V read once per WG ----
  // wave = {half, coltile}: halves split the k range, then LDS-reduce.
  const int half = wave >> 2, ct = wave & 3;
  const int nb   = ct * 16 + m;
  const int kbeg = half * (SEQ / 2), kend = kbeg + SEQ / 2;
  v8f acc0 = {}, acc1 = {};
  for (int kk0 = kbeg; kk0 < kend; kk0 += 64) {
    if (kk0 + 128 < kend)
      __builtin_prefetch(Vh + (size_t)(kk0 + 128) * HDIM + nb, 0, 1);
    for (int kk2 = 0; kk2 < 64; kk2 += 4) {
      const int kk = kk0 + kk2;
      v2f bb;
      bb.x = Vh[(size_t)(kk + koff) * HDIM + nb];
      bb.y = Vh[(size_t)(kk + koff + 1) * HDIM + nb];
      v2f a0 = *(const v2f*)(&lds_p[m * PSTR + kk + koff]);
      v2f a1 = *(const v2f*)(&lds_p[(m + 16) * PSTR + kk + koff]);
      acc0 = wmma_f32(a0, bb, acc0);
      acc1 = wmma_f32(a1, bb, acc1);
    }
  }
  float* pbuf = lds_q;  // reuse (scores phase done): 4*2*256 = 2048 floats
  if (half == 1) {
#pragma unroll
    for (int r = 0; r < 8; ++r) {
      pbuf[ct * 512 +       (rbase + r) * 16 + m] = acc0[r];
      pbuf[ct * 512 + 256 + (rbase + r) * 16 + m] = acc1[r];
    }
  }
  __syncthreads();
  if (half == 0) {
#pragma unroll
    for (int r = 0; r < 8; ++r) {
      acc0[r] += pbuf[ct * 512 +       (rbase + r) * 16 + m];
      acc1[r] += pbuf[ct * 512 + 256 + (rbase + r) * 16 + m];
    }
    const size_t obase = ((size_t)b * SEQ + qblk * 32) * DMODEL + h * HDIM + nb;
#pragma unroll
    for (int r = 0; r < 8; ++r) {
      ctx[obase + (size_t)(rbase + r) * DMODEL]      = acc0[r];
      ctx[obase + (size_t)(16 + rbase + r) * DMODEL] = acc1[r];
    }
  }
}

// =====================================================================
extern "C" void kernel_launch(void* const* d_in, const int* in_sizes, int n_in,
                              void* d_out, int out_size, void* d_ws,
                              size_t ws_size, hipStream_t stream) {
  const float* q    = (const float*)d_in[0];
  const float* k    = (const float*)d_in[1];
  const float* v    = (const float*)d_in[2];
  const float* mask = (const float*)d_in[3];
  const float* Wq   = (const float*)d_in[4];
  const float* bq   = (const float*)d_in[5];
  const float* Wk   = (const float*)d_in[6];
  const float* bk   = (const float*)d_in[7];
  const float* Wv   = (const float*)d_in[8];
  const float* bv   = (const float*)d_in[9];
  const float* Wo   = (const float*)d_in[10];
  const float* bo   = (const float*)d_in[11];

  float* out     = (float*)d_out;                       // [B,S,768]
  float* weights = out + (size_t)BATCH * SEQ * DMODEL;  // [B,H,S,S]

  const size_t headElems = (size_t)BATCH * HEADS * SEQ * HDIM;
  float* ws_q   = (float*)d_ws;
  float* ws_k   = ws_q + headElems;
  float* ws_v   = ws_k + headElems;
  float* ws_ctx = ws_v + headElems;                     // [B,S,768]

  dim3 pgrid(BATCH * SEQ / 64, DMODEL / 128);
  size_t psmem = (size_t)64 * APSTR * sizeof(float);
  mha_proj_kernel<<<pgrid, 256, psmem, stream>>>(q, Wq, bq, ws_q, 1);
  mha_proj_kernel<<<pgrid, 256, psmem, stream>>>(k, Wk, bk, ws_k, 1);
  mha_proj_kernel<<<pgrid, 256, psmem, stream>>>(v, Wv, bv, ws_v, 1);

  size_t asmem =
      (size_t)(32 * PSTR + 32 * QSTR + SEQ + 32 * 8) * sizeof(float);
  mha_attn_kernel<<<BATCH * HEADS * (SEQ / 32), 256, asmem, stream>>>(
      ws_q, ws_k, ws_v, mask, weights, ws_ctx);

  mha_proj_kernel<<<pgrid, 256, psmem, stream>>>(ws_ctx, Wo, bo, out, 0);
}